// ScaleProcess_20899310863139
// MI455X (gfx1250) — compile-verified
//
#include <hip/hip_runtime.h>
#include <hip/hip_bf16.h>

// Problem constants (match the reference)
#define NUM_BINS 128
#define NB       128            // batch
#define IMG_ELEM (512 * 512)    // 262144 floats per image
#define IMG_VEC  (IMG_ELEM / 4) // 65536 float4 per image

typedef __attribute__((ext_vector_type(2))) float v2f;
typedef __attribute__((ext_vector_type(4))) float v4f;
typedef __attribute__((ext_vector_type(8))) float v8f;

// ---------------------------------------------------------------------------
// Kernel 0: zero the per-image histogram workspace (harness poisons d_ws).
// ---------------------------------------------------------------------------
__global__ __launch_bounds__(256) void hist_zero_kernel(unsigned int* __restrict__ hist) {
    int i = blockIdx.x * blockDim.x + threadIdx.x;
    if (i < NB * NUM_BINS) hist[i] = 0u;
}

// ---------------------------------------------------------------------------
// Kernel 1: per-image histogram. Streaming float4 loads with default RT
// temporal hint (deliberately: this pass pre-loads x into the 192MB L2 so
// the scale pass re-reads hit in L2). LDS-privatized 128-bin histogram via
// ds_add atomics, one global atomic add per bin per block.
// ---------------------------------------------------------------------------
__global__ __launch_bounds__(256) void hist_kernel(const float* __restrict__ x,
                                                   unsigned int* __restrict__ hist) {
    __shared__ unsigned int lh[NUM_BINS];
    const int img = blockIdx.y;

    for (int i = threadIdx.x; i < NUM_BINS; i += blockDim.x) lh[i] = 0u;
    __syncthreads();

    const v4f* __restrict__ xi = (const v4f*)(x + (size_t)img * IMG_ELEM);
    const int stride = blockDim.x * gridDim.x;

    for (int i = blockIdx.x * blockDim.x + threadIdx.x; i < IMG_VEC; i += stride) {
        v4f v = xi[i];
#pragma unroll
        for (int j = 0; j < 4; ++j) {
            float f = v[j];
            if (f >= 0.0f && f <= 1.0f) {
                int b = (int)floorf(f * (float)NUM_BINS);
                b = b > (NUM_BINS - 1) ? (NUM_BINS - 1) : b;
                atomicAdd(&lh[b], 1u);           // ds_add_u32
            }
        }
    }
    __syncthreads();

    for (int i = threadIdx.x; i < NUM_BINS; i += blockDim.x) {
        unsigned int c = lh[i];
        if (c) atomicAdd(&hist[img * NUM_BINS + i], c);  // global_atomic_add_u32
    }
}

// ---------------------------------------------------------------------------
// Kernel 2: MLP on the histograms via V_WMMA_F32_16X16X4_F32.
//   H[128x128] (f32 counts) @ W1[128x16]  -> pre[128x16]
//   relu(pre + b1) @ W2[16x1] + b2        -> scale[128]
// One block, 8 waves; wave w owns batch rows [16w, 16w+16), accumulating
// K=128 as 32 chained 16x16x4 WMMA ops.
//
// A-fragment layout (32-bit A 16x4, ISA 7.12.2): lane l<16 -> M=l, {K0,K1};
// lane l>=16 -> M=l-16, {K2,K3}. B rows striped across lanes within a VGPR
// with the same half-wave K split. D: VGPR r, lanes 0-15 -> M=r, N=lane;
// lanes 16-31 -> M=8+r, N=lane-16.
// ---------------------------------------------------------------------------
__global__ __launch_bounds__(256) void mlp_wmma_kernel(const unsigned int* __restrict__ hist,
                                                       const float* __restrict__ W1,
                                                       const float* __restrict__ b1,
                                                       const float* __restrict__ W2,
                                                       const float* __restrict__ b2,
                                                       float* __restrict__ scale) {
    const int wave = threadIdx.x >> 5;   // 0..7 -> batch tile
    const int lane = threadIdx.x & 31;
    const int half = lane >> 4;          // 0: K0/K1 half, 1: K2/K3 half
    const int l16  = lane & 15;

    const unsigned int* __restrict__ hrow = hist + (wave * 16 + l16) * NUM_BINS;

    v8f acc = {};
#pragma unroll
    for (int k0 = 0; k0 < NUM_BINS; k0 += 4) {
        const int kb = k0 + half * 2;
        v2f a, b;
        a.x = (float)hrow[kb];
        a.y = (float)hrow[kb + 1];
        b.x = W1[(kb)     * 16 + l16];
        b.y = W1[(kb + 1) * 16 + l16];
        // v_wmma_f32_16x16x4_f32: (neg_a, A, neg_b, B, c_mod, C, reuse_a, reuse_b)
        acc = __builtin_amdgcn_wmma_f32_16x16x4_f32(
            false, a, false, b, (short)0, acc, false, false);
    }

    // Epilogue: bias + ReLU + project with W2, reduce over N (16 lanes/half).
    const float bias1 = b1[l16];
    const float w2n   = W2[l16];
    const float bias2 = b2[0];

#pragma unroll
    for (int r = 0; r < 8; ++r) {
        float h = acc[r] + bias1;
        h = h > 0.0f ? h : 0.0f;
        float c = h * w2n;
        // butterfly sum within each 16-lane half (N dimension)
        c += __shfl_xor(c, 1, 16);
        c += __shfl_xor(c, 2, 16);
        c += __shfl_xor(c, 4, 16);
        c += __shfl_xor(c, 8, 16);
        if (l16 == 0) {
            const int batch = wave * 16 + half * 8 + r;
            scale[batch] = c + bias2;
        }
    }
}

// ---------------------------------------------------------------------------
// Kernel 3: out = x * scale[img]. Streaming float4 pass #2.
//  - x loads: non-temporal (last use; they hit in L2 from pass #1, no need to
//    retain afterwards).
//  - out stores: non-temporal so the 134MB output stream does NOT allocate in
//    L2 and evict the still-needed tail of x (x 134MB + out 134MB > 192MB L2).
// ---------------------------------------------------------------------------
__global__ __launch_bounds__(256) void scale_kernel(const float* __restrict__ x,
                                                    const float* __restrict__ scale,
                                                    float* __restrict__ out) {
    const int img = blockIdx.y;
    const float s = scale[img];
    const size_t base = (size_t)img * IMG_VEC;
    const int i = blockIdx.x * blockDim.x + threadIdx.x;
    if (i < IMG_VEC) {
        const v4f* xp = (const v4f*)x + base + i;
        v4f v = __builtin_nontemporal_load(xp);
        v *= s;
        v4f* op = (v4f*)out + base + i;
        __builtin_nontemporal_store(v, op);
    }
}

// ---------------------------------------------------------------------------
extern "C" void kernel_launch(void* const* d_in, const int* in_sizes, int n_in,
                              void* d_out, int out_size, void* d_ws, size_t ws_size,
                              hipStream_t stream) {
    const float* x  = (const float*)d_in[0];
    const float* W1 = (const float*)d_in[1];
    const float* b1 = (const float*)d_in[2];
    const float* W2 = (const float*)d_in[3];
    const float* b2 = (const float*)d_in[4];
    float* out = (float*)d_out;

    unsigned int* hist = (unsigned int*)d_ws;                          // 128*128 u32 = 64 KB
    float* scale       = (float*)((char*)d_ws + NB * NUM_BINS * 4);    // 128 f32

    // 0) zero histograms
    hist_zero_kernel<<<(NB * NUM_BINS + 255) / 256, 256, 0, stream>>>(hist);

    // 1) per-image histograms: 32 blocks x 256 threads per image
    hist_kernel<<<dim3(32, NB), 256, 0, stream>>>(x, hist);

    // 2) WMMA MLP -> per-image scales (one block, 8 waves)
    mlp_wmma_kernel<<<1, 256, 0, stream>>>(hist, W1, b1, W2, b2, scale);

    // 3) scale pass (NT loads of x, NT stores of out)
    scale_kernel<<<dim3(IMG_VEC / 256, NB), 256, 0, stream>>>(x, scale, out);
}